// CoralFocalLoss_MultiTask_25305947308583
// MI455X (gfx1250) — compile-verified
//
#include <hip/hip_runtime.h>

// CORAL focal loss, 3 tasks fused into one bandwidth-bound pass.
// Stage 1: 1024 blocks x 256 threads, grid-stride over N samples, per-block
//          partial sums (kl, jsnm, jsnl) via WMMA wave32 reduction + LDS.
// Stage 2: single block reduces the 1024 partials and writes the 4 outputs:
//          [total, l_kl, l_jsnm, l_jsnl].

#define NBLOCKS  1024
#define NTHREADS 256
#define NWAVES   (NTHREADS / 32)

typedef __attribute__((ext_vector_type(2))) float v2f;
typedef __attribute__((ext_vector_type(8))) float v8f;

// Full wave32 sum using V_WMMA_F32_16X16X4_F32.
// A (16x4, documented layout): lane L<16 holds A[L][0],A[L][1]; lane L>=16
// holds A[L-16][2],A[L-16][3]. With a={s,0}: row m of A = {s_m,0,s_{m+16},0}.
// B = all ones (layout independent) => D[m][n] = s_m + s_{m+16} for every n.
// D VGPRs 0..7: lanes<16 carry rows 0..7, lanes>=16 carry rows 8..15, so
// sum(c[0..7]) gives half the total per lane-half; xor-16 combines them.
__device__ __forceinline__ float wave_sum32(float s) {
#if __has_builtin(__builtin_amdgcn_wmma_f32_16x16x4_f32)
  v2f a; a[0] = s;    a[1] = 0.0f;
  v2f b; b[0] = 1.0f; b[1] = 1.0f;
  v8f c = {};
  c = __builtin_amdgcn_wmma_f32_16x16x4_f32(
      /*neg_a=*/false, a, /*neg_b=*/false, b,
      /*c_mod=*/(short)0, c, /*reuse_a=*/false, /*reuse_b=*/false);
  float h = ((c[0] + c[1]) + (c[2] + c[3])) + ((c[4] + c[5]) + (c[6] + c[7]));
  h += __shfl_xor(h, 16, 32);
  return h;
#else
  for (int off = 16; off > 0; off >>= 1) s += __shfl_xor(s, off, 32);
  return s;
#endif
}

// One CORAL-focal element: tm = (col < target).
// pt = sigmoid(tm ? x : -x); alpha = tm?0.25:0.75; GAMMA=2 -> (1-pt)^2.
__device__ __forceinline__ float elem_loss(float x, bool tm) {
  float y = tm ? x : -x;
  float q = 1.0f / (1.0f + __expf(-y));          // pt
  float a = tm ? 0.25f : 0.75f;                  // alpha_t
  float omq = 1.0f - q;                          // focal base
  return -a * omq * omq * __logf(q + 1e-8f);
}

__global__ void coral_focal_partial(const float* __restrict__ kl_logits,
                                    const float* __restrict__ jm_logits,
                                    const float* __restrict__ jl_logits,
                                    const float* __restrict__ cw,
                                    const int* __restrict__ kl_t,
                                    const int* __restrict__ jm_t,
                                    const int* __restrict__ jl_t,
                                    float* __restrict__ partials, int n) {
  const int tid    = blockIdx.x * blockDim.x + threadIdx.x;
  const int stride = gridDim.x * blockDim.x;

  // Keep the 5 class weights in registers; per-lane branchless select
  // instead of a gather through the cache every sample.
  const float c0 = cw[0], c1 = cw[1], c2 = cw[2], c3 = cw[3], c4 = cw[4];

  float s_kl = 0.0f, s_jm = 0.0f, s_jl = 0.0f;

  for (int i = tid; i < n; i += stride) {
    const int tk = kl_t[i];
    float w = c0;
    w = (tk == 1) ? c1 : w;
    w = (tk == 2) ? c2 : w;
    w = (tk == 3) ? c3 : w;
    w = (tk == 4) ? c4 : w;

    // [N,4] tensor: one 128-bit coalesced load per sample.
    const float4 xk = reinterpret_cast<const float4*>(kl_logits)[i];
    float akl = elem_loss(xk.x, 0 < tk) + elem_loss(xk.y, 1 < tk) +
                elem_loss(xk.z, 2 < tk) + elem_loss(xk.w, 3 < tk);
    s_kl += w * akl;

    const int tm = jm_t[i];
    const float* pm = jm_logits + 3 * (size_t)i;
    s_jm += w * (elem_loss(pm[0], 0 < tm) + elem_loss(pm[1], 1 < tm) +
                 elem_loss(pm[2], 2 < tm));

    const int tl = jl_t[i];
    const float* pl = jl_logits + 3 * (size_t)i;
    s_jl += w * (elem_loss(pl[0], 0 < tl) + elem_loss(pl[1], 1 < tl) +
                 elem_loss(pl[2], 2 < tl));
  }

  // All lanes converged here -> EXEC all ones, legal for WMMA.
  s_kl = wave_sum32(s_kl);
  s_jm = wave_sum32(s_jm);
  s_jl = wave_sum32(s_jl);

  __shared__ float lds[3][NWAVES];
  const int lane = threadIdx.x & 31;
  const int wid  = threadIdx.x >> 5;
  if (lane == 0) {
    lds[0][wid] = s_kl;
    lds[1][wid] = s_jm;
    lds[2][wid] = s_jl;
  }
  __syncthreads();
  if (threadIdx.x == 0) {
    float a = 0.0f, b = 0.0f, d = 0.0f;
    for (int w2 = 0; w2 < NWAVES; ++w2) {
      a += lds[0][w2]; b += lds[1][w2]; d += lds[2][w2];
    }
    partials[3 * blockIdx.x + 0] = a;
    partials[3 * blockIdx.x + 1] = b;
    partials[3 * blockIdx.x + 2] = d;
  }
}

__global__ void coral_focal_final(const float* __restrict__ partials,
                                  float* __restrict__ out, int nblocks,
                                  float n) {
  float s0 = 0.0f, s1 = 0.0f, s2 = 0.0f;
  for (int i = threadIdx.x; i < nblocks; i += blockDim.x) {
    s0 += partials[3 * i + 0];
    s1 += partials[3 * i + 1];
    s2 += partials[3 * i + 2];
  }
  s0 = wave_sum32(s0);
  s1 = wave_sum32(s1);
  s2 = wave_sum32(s2);

  __shared__ float lds[3][NWAVES];
  const int lane = threadIdx.x & 31;
  const int wid  = threadIdx.x >> 5;
  if (lane == 0) {
    lds[0][wid] = s0; lds[1][wid] = s1; lds[2][wid] = s2;
  }
  __syncthreads();
  if (threadIdx.x == 0) {
    float a = 0.0f, b = 0.0f, c = 0.0f;
    for (int w = 0; w < NWAVES; ++w) {
      a += lds[0][w]; b += lds[1][w]; c += lds[2][w];
    }
    const float l_kl = a / (n * 4.0f);
    const float l_jm = b / (n * 3.0f);
    const float l_jl = c / (n * 3.0f);
    out[0] = (l_kl + l_jm + l_jl) / 3.0f;  // total
    out[1] = l_kl;
    out[2] = l_jm;
    out[3] = l_jl;
  }
}

extern "C" void kernel_launch(void* const* d_in, const int* in_sizes, int n_in,
                              void* d_out, int out_size, void* d_ws,
                              size_t ws_size, hipStream_t stream) {
  const float* kl_logits = (const float*)d_in[0];  // [N,4] f32
  const float* jm_logits = (const float*)d_in[1];  // [N,3] f32
  const float* jl_logits = (const float*)d_in[2];  // [N,3] f32
  const float* cw        = (const float*)d_in[3];  // [5]   f32
  const int*   kl_t      = (const int*)d_in[4];    // [N]   i32
  const int*   jm_t      = (const int*)d_in[5];    // [N]   i32
  const int*   jl_t      = (const int*)d_in[6];    // [N]   i32

  const int n = in_sizes[4];               // N samples
  float* partials = (float*)d_ws;          // 3 * NBLOCKS floats (12 KB)

  coral_focal_partial<<<NBLOCKS, NTHREADS, 0, stream>>>(
      kl_logits, jm_logits, jl_logits, cw, kl_t, jm_t, jl_t, partials, n);
  coral_focal_final<<<1, NTHREADS, 0, stream>>>(partials, (float*)d_out,
                                                NBLOCKS, (float)n);
}